// MambaModel_53085795778726
// MI455X (gfx1250) — compile-verified
//
#include <hip/hip_runtime.h>
#include <hip/hip_bf16.h>

// ---- problem constants (match reference) ----
#define BB   2
#define LL   2048
#define DD   256
#define DI   512          // E*D
#define XZW  1024         // 2*DI
#define DS   16
#define DC   4
#define DTR  16
#define DBLW 48           // DTR + 2*DS
#define NL   4
#define EPS  1e-5f

typedef float v2f __attribute__((ext_vector_type(2)));
typedef float v8f __attribute__((ext_vector_type(8)));

__device__ __forceinline__ float siluf(float x) {
  return x / (1.f + __expf(-x));
}
__device__ __forceinline__ float softplusf(float x) {
  return x > 20.f ? x : log1pf(__expf(x));
}

// ---------------------------------------------------------------------------
// (B, D, L) -> (B, L, D)
// ---------------------------------------------------------------------------
__global__ void transpose_src_kernel(const float* __restrict__ src,
                                     float* __restrict__ h) {
  int idx = blockIdx.x * blockDim.x + threadIdx.x;   // over B*L*D
  int d = idx % DD;
  int t = idx / DD;
  int l = t % LL;
  int b = t / LL;
  h[idx] = src[((size_t)b * DD + d) * LL + l];
}

// ---------------------------------------------------------------------------
// row-wise RMSNorm over D=256: one block (256 thr) per row
// ---------------------------------------------------------------------------
__global__ void rmsnorm_kernel(const float* __restrict__ x,
                               const float* __restrict__ w,
                               float* __restrict__ out) {
  __shared__ float sred[8];
  const int row = blockIdx.x;
  const int d   = threadIdx.x;
  float v = x[(size_t)row * DD + d];
  float ss = v * v;
  #pragma unroll
  for (int off = 16; off; off >>= 1) ss += __shfl_xor(ss, off, 32);
  if ((threadIdx.x & 31) == 0) sred[threadIdx.x >> 5] = ss;
  __syncthreads();
  float sum = 0.f;
  #pragma unroll
  for (int i = 0; i < 8; i++) sum += sred[i];
  float scale = rsqrtf(sum * (1.f / DD) + EPS);
  out[(size_t)row * DD + d] = v * scale * w[d];
}

// ---------------------------------------------------------------------------
// Generic FP32 WMMA GEMM:  Out[M,N] = A[M,KC](lda) * W[N,KC]^T  (+ epilogue)
//   MODE 0: plain store
//   MODE 1: residual: C operand preloaded from aux[M,N]; D written to Out
//   MODE 2: out = softplus(acc + aux[n])   (per-column bias)
//   STAGEA: stage the block-shared 16-row A tile into LDS with
//           global_load_async_to_lds_b128 (+ s_wait_asynccnt), then feed the
//           WMMA A-fragments from LDS (row pad +4 floats -> conflict-free
//           ds_load_b64). Requires grid = (tilesN/8, tilesM).
// One wave computes one 16x16 tile via V_WMMA_F32_16X16X4_F32, K step 4.
// FP32 fragment layouts (ISA 7.12.2): A lane(m,half): {A[m][k+2h], A[m][k+2h+1]}
//                                     B lane(n,half): {W[n][k+2h], W[n][k+2h+1]}
//                                     C vgpr r lane(n,half): C[r+8h][n]
// ---------------------------------------------------------------------------
template <int MODE, int KC, bool STAGEA>
__global__ void gemm_wmma_f32_kernel(const float* __restrict__ A, int lda,
                                     const float* __restrict__ W,
                                     float* __restrict__ Out,
                                     const float* __restrict__ aux,
                                     int M, int N) {
  constexpr int KP = KC + 4;                 // padded LDS row stride (floats)
  __shared__ float As[STAGEA ? 16 * KP : 4];

  const int lane = threadIdx.x & 31;
  const int wave = threadIdx.x >> 5;
  const int half = lane >> 4;
  const int mn   = lane & 15;

  int tm, tn;
  if (STAGEA) {
    tm = blockIdx.y;
    tn = blockIdx.x * (blockDim.x >> 5) + wave;
  } else {
    const int tilesN = N >> 4;
    const int tile = blockIdx.x * (blockDim.x >> 5) + wave;
    tm = tile / tilesN;
    tn = tile - tm * tilesN;
  }

  if (STAGEA) {
    // async-copy the 16 x KC A tile (row stride lda) into LDS
    const int rowChunks = KC >> 2;                   // 16B chunks per row
    const int chunks = 16 * rowChunks;
    const float* Abase = A + (size_t)(tm * 16) * lda;
    for (int c = threadIdx.x; c < chunks; c += blockDim.x) {
      const int r  = c / rowChunks;
      const int kk = (c - r * rowChunks) << 2;
      const float* gp = Abase + (size_t)r * lda + kk;
      unsigned lds_off = (unsigned)(size_t)(&As[r * KP + kk]);
      unsigned long long ga = (unsigned long long)(size_t)gp;
      asm volatile("global_load_async_to_lds_b128 %0, %1, off"
                   :: "v"(lds_off), "v"(ga) : "memory");
    }
    asm volatile("s_wait_asynccnt 0x0" ::: "memory");
    __syncthreads();
  }

  const float* arow = A + (size_t)(tm * 16 + mn) * lda;
  const float* wrow = W + (size_t)(tn * 16 + mn) * KC;

  v8f c;
  if (MODE == 1) {
    const float* r = aux + (size_t)(tm * 16) * N + tn * 16;
    #pragma unroll
    for (int i = 0; i < 8; i++) c[i] = r[(size_t)(i + 8 * half) * N + mn];
  } else {
    #pragma unroll
    for (int i = 0; i < 8; i++) c[i] = 0.f;
  }

  for (int k = 0; k < KC; k += 4) {
    v2f a;
    if (STAGEA) a = *(const v2f*)(&As[mn * KP + k + 2 * half]);
    else        a = *(const v2f*)(arow + k + 2 * half);
    v2f b = *(const v2f*)(wrow + k + 2 * half);
    c = __builtin_amdgcn_wmma_f32_16x16x4_f32(false, a, false, b,
                                              (short)0, c, false, false);
  }

  float* orow = Out + (size_t)(tm * 16) * N + tn * 16;
  float bias = (MODE == 2) ? aux[tn * 16 + mn] : 0.f;
  #pragma unroll
  for (int i = 0; i < 8; i++) {
    float v = c[i];
    if (MODE == 2) v = softplusf(v + bias);
    orow[(size_t)(i + 8 * half) * N + mn] = v;
  }
}

// ---------------------------------------------------------------------------
// depthwise causal conv (DC=4) over x = xz[..., :DI], + bias, + SiLU
// ---------------------------------------------------------------------------
__global__ void conv_silu_kernel(const float* __restrict__ xz,
                                 const float* __restrict__ cw,
                                 const float* __restrict__ cb,
                                 float* __restrict__ xc) {
  int idx = blockIdx.x * blockDim.x + threadIdx.x;   // over B*L*DI
  int d = idx % DI;
  int t = idx / DI;
  int l = t % LL;
  int b = t / LL;
  float acc = cb[d];
  #pragma unroll
  for (int j = 0; j < DC; j++) {
    int ls = l - (DC - 1) + j;
    if (ls >= 0) {
      acc += xz[((size_t)b * LL + ls) * XZW + d] * cw[d * DC + j];
    }
  }
  xc[idx] = siluf(acc);
}

// ---------------------------------------------------------------------------
// selective scan: one 16-lane half-wave per (b, channel d); lane = state n.
//   h = h*exp(delta*A[d,n]) + delta*B[l,n]*x ;  y = sum_n h*C[l,n]
// fused: + D_param*x, gate with silu(z)
// Software-pipelined: next-step operands load while exp/fma of this step run;
// global_prefetch pulls rows 64 steps ahead.
// ---------------------------------------------------------------------------
__global__ void scan_kernel(const float* __restrict__ delta,
                            const float* __restrict__ xc,
                            const float* __restrict__ dbl,
                            const float* __restrict__ xz,
                            const float* __restrict__ A_log,
                            const float* __restrict__ Dp,
                            float* __restrict__ ybuf) {
  const int lane = threadIdx.x & 31;
  const int wave = threadIdx.x >> 5;
  const int half = lane >> 4;
  const int n    = lane & 15;
  const int chan = (blockIdx.x * (blockDim.x >> 5) + wave) * 2 + half; // 0..B*DI-1
  const int b = chan / DI;
  const int d = chan - b * DI;

  const float Acoef = -__expf(A_log[d * DS + n]);
  const float Dcoef = Dp[d];
  float h = 0.f;
  const size_t rbase = (size_t)b * LL;

  size_t row = rbase;
  float dl = delta[row * DI + d];
  float xv = xc[row * DI + d];
  float Bn = dbl[row * DBLW + DTR + n];
  float Cn = dbl[row * DBLW + DTR + DS + n];

  for (int l = 0; l < LL; l++) {
    float e   = __expf(dl * Acoef);      // TRANS, off the h-chain
    float dbu = dl * Bn * xv;

    // issue next-iteration loads before touching the dependent chain
    float dl2 = 0.f, xv2 = 0.f, Bn2 = 0.f, Cn2 = 0.f;
    if (l + 1 < LL) {
      const size_t r2 = row + 1;
      dl2 = delta[r2 * DI + d];
      xv2 = xc[r2 * DI + d];
      Bn2 = dbl[r2 * DBLW + DTR + n];
      Cn2 = dbl[r2 * DBLW + DTR + DS + n];
    }
    if ((l & 63) == 0 && l + 64 < LL) {
      const size_t rp = row + 64;
      __builtin_prefetch(&delta[rp * DI + d], 0, 0);
      __builtin_prefetch(&xc[rp * DI + d], 0, 0);
      __builtin_prefetch(&dbl[rp * DBLW + DTR + n], 0, 0);
      __builtin_prefetch(&xz[rp * XZW + DI + d], 0, 0);
    }

    h = fmaf(h, e, dbu);                 // the only serial dependency
    float y = h * Cn;
    #pragma unroll
    for (int off = 8; off; off >>= 1) y += __shfl_xor(y, off, 16);
    if (n == 0) {
      float zv = xz[row * XZW + DI + d];
      ybuf[row * DI + d] = (y + Dcoef * xv) * siluf(zv);
    }

    dl = dl2; xv = xv2; Bn = Bn2; Cn = Cn2;
    row++;
  }
}

// ---------------------------------------------------------------------------
// final: gather row lengths[b]-1, RMSNorm with normf_w -> out[b, :]
// ---------------------------------------------------------------------------
__global__ void final_kernel(const float* __restrict__ h,
                             const int* __restrict__ lengths,
                             const float* __restrict__ wf,
                             float* __restrict__ out) {
  __shared__ float sred[8];
  const int b = blockIdx.x;
  const int d = threadIdx.x;
  int row = lengths[b] - 1;
  row = row < 0 ? 0 : (row >= LL ? LL - 1 : row);
  float v = h[((size_t)b * LL + row) * DD + d];
  float ss = v * v;
  #pragma unroll
  for (int off = 16; off; off >>= 1) ss += __shfl_xor(ss, off, 32);
  if ((threadIdx.x & 31) == 0) sred[threadIdx.x >> 5] = ss;
  __syncthreads();
  float sum = 0.f;
  #pragma unroll
  for (int i = 0; i < 8; i++) sum += sred[i];
  float scale = rsqrtf(sum * (1.f / DD) + EPS);
  out[b * DD + d] = v * scale * wf[d];
}

// ---------------------------------------------------------------------------
extern "C" void kernel_launch(void* const* d_in, const int* in_sizes, int n_in,
                              void* d_out, int out_size, void* d_ws, size_t ws_size,
                              hipStream_t stream) {
  const float* src       = (const float*)d_in[0];
  const int*   lengths   = (const int*)  d_in[1];
  const float* norm_w    = (const float*)d_in[2];
  const float* in_proj_w = (const float*)d_in[3];
  const float* conv_w    = (const float*)d_in[4];
  const float* conv_b    = (const float*)d_in[5];
  const float* x_proj_w  = (const float*)d_in[6];
  const float* dt_proj_w = (const float*)d_in[7];
  const float* dt_proj_b = (const float*)d_in[8];
  const float* A_log     = (const float*)d_in[9];
  const float* D_param   = (const float*)d_in[10];
  const float* out_proj_w= (const float*)d_in[11];
  const float* normf_w   = (const float*)d_in[12];
  float* out = (float*)d_out;

  // ---- workspace layout (floats); total ~12.8M floats (~49 MB) ----
  float* ws = (float*)d_ws;
  const size_t BLD  = (size_t)BB * LL * DD;   // 1,048,576
  const size_t BLDI = (size_t)BB * LL * DI;   // 2,097,152
  float* hbuf  = ws;                 ws += BLD;
  float* hn    = ws;                 ws += BLD;
  float* xz    = ws;                 ws += (size_t)BB * LL * XZW;
  float* xc    = ws;                 ws += BLDI;
  float* dbl   = ws;                 ws += (size_t)BB * LL * DBLW;
  float* delta = ws;                 ws += BLDI;
  float* ybuf  = ws;                 ws += BLDI;
  (void)ws_size; (void)in_sizes; (void)n_in; (void)out_size;

  const int M = BB * LL;   // 4096 rows

  // (B,D,L) -> (B,L,D)
  transpose_src_kernel<<<(int)(BLD / 256), 256, 0, stream>>>(src, hbuf);

  for (int i = 0; i < NL; i++) {
    const float* nw_i  = norm_w     + (size_t)i * DD;
    const float* ipw_i = in_proj_w  + (size_t)i * XZW * DD;
    const float* cw_i  = conv_w     + (size_t)i * DI * DC;
    const float* cb_i  = conv_b     + (size_t)i * DI;
    const float* xpw_i = x_proj_w   + (size_t)i * DBLW * DI;
    const float* dpw_i = dt_proj_w  + (size_t)i * DI * DTR;
    const float* dpb_i = dt_proj_b  + (size_t)i * DI;
    const float* al_i  = A_log      + (size_t)i * DI * DS;
    const float* Dp_i  = D_param    + (size_t)i * DI;
    const float* opw_i = out_proj_w + (size_t)i * DD * DI;

    // 1) hn = rmsnorm(h) * norm_w
    rmsnorm_kernel<<<M, 256, 0, stream>>>(hbuf, nw_i, hn);

    // 2) xz = hn @ in_proj_w^T           (4096 x 1024, K=256, A tile via LDS)
    gemm_wmma_f32_kernel<0, DD, true>
        <<<dim3((XZW / 16) / 8, M / 16), 256, 0, stream>>>(
        hn, DD, ipw_i, xz, nullptr, M, XZW);

    // 3) xc = silu(causal_conv(x) + b)
    conv_silu_kernel<<<(int)(BLDI / 256), 256, 0, stream>>>(xz, cw_i, cb_i, xc);

    // 4) dbl = xc @ x_proj_w^T           (4096 x 48, K=512; tilesN=3 -> direct)
    gemm_wmma_f32_kernel<0, DI, false>
        <<<(M / 16) * (DBLW / 16) / 8, 256, 0, stream>>>(
        xc, DI, xpw_i, dbl, nullptr, M, DBLW);

    // 5) delta = softplus(dt @ dt_proj_w^T + b)   (A = dbl[:, :16], lda=48)
    gemm_wmma_f32_kernel<2, DTR, true>
        <<<dim3((DI / 16) / 8, M / 16), 256, 0, stream>>>(
        dbl, DBLW, dpw_i, delta, dpb_i, M, DI);

    // 6) selective scan + D*x skip + silu(z) gate -> ybuf
    scan_kernel<<<(BB * DI / 2) / 8, 256, 0, stream>>>(
        delta, xc, dbl, xz, al_i, Dp_i, ybuf);

    // 7) h = h + ybuf @ out_proj_w^T   (residual rides the WMMA C operand)
    gemm_wmma_f32_kernel<1, DI, true>
        <<<dim3((DD / 16) / 8, M / 16), 256, 0, stream>>>(
        ybuf, DI, opw_i, hbuf, hbuf, M, DD);
  }

  // final rmsnorm on only the two gathered rows
  final_kernel<<<BB, 256, 0, stream>>>(hbuf, lengths, normf_w, out);
}